// SharedGATv2Layer_54322746360398
// MI455X (gfx1250) — compile-verified
//
#include <hip/hip_runtime.h>
#include <hip/hip_bf16.h>

#define N_NODES   50000
#define N_EDGES   800000
#define E_TOT     850000     // edges + self loops
#define IN_F      128
#define HEADS     4
#define OUT_F     32
#define HF        128        // HEADS*OUT_F
#define NEG_SLOPE 0.2f

typedef __attribute__((ext_vector_type(2))) float v2f;
typedef __attribute__((ext_vector_type(8))) float v8f;

// ---------------------------------------------------------------------------
// Kernel 0: init output with bias broadcast; init softmax stats
// stats[0..3] = running max per head, stats[4..7] = running sum(exp) per head
// ---------------------------------------------------------------------------
__global__ void gat_init(float* __restrict__ out, const float* __restrict__ bias,
                         float* __restrict__ stats) {
    long i = (long)blockIdx.x * blockDim.x + threadIdx.x;
    if (i < (long)N_NODES * HF) out[i] = bias[i & (HF - 1)];
    if (i < 4)      stats[i] = -3.402823466e38f;
    else if (i < 8) stats[i] = 0.0f;
}

// ---------------------------------------------------------------------------
// Kernel 1: h = x @ W^T via V_WMMA_F32_16X16X4_F32  (wave32)
//   block = 256 threads = 8 waves; block b -> rows [16b,16b+16),
//   wave w -> cols [16w,16w+16). 50000/16 = 3125 blocks exactly -> EXEC all-1s.
// A (16x4 f32): lanes 0-15 M=lane, VGPR{0,1}=K{0,1}; lanes 16-31 K{2,3}.
// B (4x16 f32): lanes 0-15 N=lane, VGPR{0,1}=K{0,1}; lanes 16-31 K{2,3}.
//   B[k][n] = W[n][k] since h[m][n] = sum_k x[m][k]*W[n][k].
// C (16x16 f32): VGPR v = row (v + 8*half), col = lane&15.
// ---------------------------------------------------------------------------
__global__ __launch_bounds__(256) void gat_gemm_wmma(const float* __restrict__ x,
                                                     const float* __restrict__ W,
                                                     float* __restrict__ hfeat) {
    const int wave = threadIdx.x >> 5;
    const int lane = threadIdx.x & 31;
    const int half = lane >> 4;
    const int l16  = lane & 15;
    const int row0 = blockIdx.x * 16;
    const int col0 = wave * 16;

    const float* __restrict__ xrow = x + (size_t)(row0 + l16) * IN_F;   // A row M=l16
    const float* __restrict__ wrow = W + (size_t)(col0 + l16) * IN_F;   // B col N=l16

    v8f acc = {};
#pragma unroll
    for (int k0 = 0; k0 < IN_F; k0 += 4) {
        const int k = k0 + 2 * half;
        v2f a, b;
        a[0] = xrow[k]; a[1] = xrow[k + 1];
        b[0] = wrow[k]; b[1] = wrow[k + 1];
        acc = __builtin_amdgcn_wmma_f32_16x16x4_f32(
            /*neg_a=*/false, a, /*neg_b=*/false, b,
            /*c_mod=*/(short)0, acc, /*reuse_a=*/false, /*reuse_b=*/false);
    }
#pragma unroll
    for (int v = 0; v < 8; ++v)
        hfeat[(size_t)(row0 + v + 8 * half) * HF + col0 + l16] = acc[v];
}

// ---------------------------------------------------------------------------
// Kernel 2: per-(node,head) attention scalars
//   a_src[n,h] = sum_f lrelu(h[n,h,f]) * att[h,f]
//   a_dst[n,h] = sum_f lrelu(h[n,h,f]) * att[h,OUT_F+f]
// (score separability: leaky_relu is elementwise, att dot splits exactly)
// ---------------------------------------------------------------------------
__global__ void gat_node_scores(const float* __restrict__ hfeat,
                                const float* __restrict__ att,
                                float* __restrict__ a_src,
                                float* __restrict__ a_dst) {
    int idx = blockIdx.x * blockDim.x + threadIdx.x;
    if (idx >= N_NODES * HEADS) return;
    const int n = idx >> 2, h = idx & 3;
    const float* __restrict__ hp = hfeat + (size_t)n * HF + h * OUT_F;
    const float* __restrict__ ap = att + h * 2 * OUT_F;
    float s0 = 0.f, s1 = 0.f;
#pragma unroll
    for (int f = 0; f < OUT_F; ++f) {
        const float v  = hp[f];
        const float lr = v > 0.f ? v : v * NEG_SLOPE;
        s0 += lr * ap[f];
        s1 += lr * ap[OUT_F + f];
    }
    a_src[idx] = s0;
    a_dst[idx] = s1;
}

__device__ __forceinline__ void atomicMaxF(float* addr, float v) {
    if (v >= 0.f) atomicMax((int*)addr, __float_as_int(v));
    else          atomicMin((unsigned int*)addr, __float_as_uint(v));
}

// ---------------------------------------------------------------------------
// Kernel 3: per-edge scores (all 4 heads per thread) + global max reduction
// ---------------------------------------------------------------------------
__global__ void gat_edge_scores_max(const int* __restrict__ ei,
                                    const float* __restrict__ a_src,
                                    const float* __restrict__ a_dst,
                                    float* __restrict__ scores,
                                    float* __restrict__ statsMax) {
    const int e = blockIdx.x * blockDim.x + threadIdx.x;
    float4 s = make_float4(-3.402823466e38f, -3.402823466e38f,
                           -3.402823466e38f, -3.402823466e38f);
    if (e < E_TOT) {
        int src, dst;
        if (e < N_EDGES) { src = ei[e]; dst = ei[N_EDGES + e]; }
        else             { src = dst = e - N_EDGES; }
        const float4 as = ((const float4*)a_src)[src];
        const float4 ad = ((const float4*)a_dst)[dst];
        s = make_float4(as.x + ad.x, as.y + ad.y, as.z + ad.z, as.w + ad.w);
        ((float4*)scores)[e] = s;
    }
    // wave32 tree max
    for (int off = 16; off > 0; off >>= 1) {
        s.x = fmaxf(s.x, __shfl_xor(s.x, off, 32));
        s.y = fmaxf(s.y, __shfl_xor(s.y, off, 32));
        s.z = fmaxf(s.z, __shfl_xor(s.z, off, 32));
        s.w = fmaxf(s.w, __shfl_xor(s.w, off, 32));
    }
    if ((threadIdx.x & 31) == 0) {
        atomicMaxF(&statsMax[0], s.x);
        atomicMaxF(&statsMax[1], s.y);
        atomicMaxF(&statsMax[2], s.z);
        atomicMaxF(&statsMax[3], s.w);
    }
}

// ---------------------------------------------------------------------------
// Kernel 4: global sum(exp(score - max)) per head
// ---------------------------------------------------------------------------
__global__ void gat_edge_sumexp(const float* __restrict__ scores,
                                const float* __restrict__ statsMax,
                                float* __restrict__ statsSum) {
    const int e = blockIdx.x * blockDim.x + threadIdx.x;
    float4 v = make_float4(0.f, 0.f, 0.f, 0.f);
    if (e < E_TOT) {
        const float4 s = ((const float4*)scores)[e];
        v.x = __expf(s.x - statsMax[0]);
        v.y = __expf(s.y - statsMax[1]);
        v.z = __expf(s.z - statsMax[2]);
        v.w = __expf(s.w - statsMax[3]);
    }
    for (int off = 16; off > 0; off >>= 1) {
        v.x += __shfl_xor(v.x, off, 32);
        v.y += __shfl_xor(v.y, off, 32);
        v.z += __shfl_xor(v.z, off, 32);
        v.w += __shfl_xor(v.w, off, 32);
    }
    if ((threadIdx.x & 31) == 0) {
        atomicAdd(&statsSum[0], v.x);
        atomicAdd(&statsSum[1], v.y);
        atomicAdd(&statsSum[2], v.z);
        atomicAdd(&statsSum[3], v.w);
    }
}

// ---------------------------------------------------------------------------
// Kernel 5: out[dst] += w * h[src]. One wave per edge; lane l handles the 4
// contiguous features 4l..4l+3 (head = l>>3). h fits in L2 -> gathers are
// L2 hits; 4 f32 atomics per lane.
// ---------------------------------------------------------------------------
__global__ __launch_bounds__(256) void gat_aggregate(const int* __restrict__ ei,
                                                     const float* __restrict__ hfeat,
                                                     const float* __restrict__ scores,
                                                     const float* __restrict__ stats,
                                                     float* __restrict__ out) {
    const int wave = threadIdx.x >> 5;
    const int lane = threadIdx.x & 31;
    const long e = (long)blockIdx.x * 8 + wave;
    if (e >= E_TOT) return;
    int src, dst;
    if (e < N_EDGES) { src = ei[e]; dst = ei[N_EDGES + e]; }
    else             { src = dst = (int)(e - N_EDGES); }
    const int h = lane >> 3;
    const float w = __expf(scores[e * 4 + h] - stats[h]) / stats[4 + h];
    const float4 hv = ((const float4*)(hfeat + (size_t)src * HF))[lane];
    float* op = out + (size_t)dst * HF + lane * 4;
    atomicAdd(op + 0, hv.x * w);
    atomicAdd(op + 1, hv.y * w);
    atomicAdd(op + 2, hv.z * w);
    atomicAdd(op + 3, hv.w * w);
}

// ---------------------------------------------------------------------------
extern "C" void kernel_launch(void* const* d_in, const int* in_sizes, int n_in,
                              void* d_out, int out_size, void* d_ws, size_t ws_size,
                              hipStream_t stream) {
    const float* x    = (const float*)d_in[0];
    const int*   ei   = (const int*)  d_in[1];
    const float* W    = (const float*)d_in[2];
    const float* att  = (const float*)d_in[3];
    const float* bias = (const float*)d_in[4];
    float* out = (float*)d_out;

    // workspace layout (floats)
    float* hfeat  = (float*)d_ws;                         // 50000*128  = 6,400,000
    float* a_src  = hfeat  + (size_t)N_NODES * HF;        // 200,000
    float* a_dst  = a_src  + (size_t)N_NODES * HEADS;     // 200,000
    float* scores = a_dst  + (size_t)N_NODES * HEADS;     // 850000*4   = 3,400,000
    float* stats  = scores + (size_t)E_TOT * HEADS;       // 8

    const long out_elems = (long)N_NODES * HF;
    gat_init<<<(unsigned)((out_elems + 255) / 256), 256, 0, stream>>>(out, bias, stats);

    gat_gemm_wmma<<<N_NODES / 16, 256, 0, stream>>>(x, W, hfeat);

    gat_node_scores<<<(N_NODES * HEADS + 255) / 256, 256, 0, stream>>>(hfeat, att, a_src, a_dst);

    gat_edge_scores_max<<<(E_TOT + 255) / 256, 256, 0, stream>>>(ei, a_src, a_dst, scores, stats);

    gat_edge_sumexp<<<(E_TOT + 255) / 256, 256, 0, stream>>>(scores, stats, stats + 4);

    gat_aggregate<<<(E_TOT + 7) / 8, 256, 0, stream>>>(ei, hfeat, scores, stats, out);
}